// EnhancedGNN_57011395887506
// MI455X (gfx1250) — compile-verified
//
#include <hip/hip_runtime.h>
#include <math.h>

#define NN 100000
#define EE 1600000
#define DD 128
#define HH 128
#define SS 64
#define GG 512
#define EPSV 1e-5f

#define A_STRIDE 132            // 128 + 4 dword pad per row (LDS bank-conflict avoidance)

typedef __attribute__((ext_vector_type(2))) float v2f;
typedef __attribute__((ext_vector_type(8))) float v8f;
typedef __attribute__((ext_vector_type(4))) unsigned int v4u;
typedef __attribute__((ext_vector_type(8))) int v8i;
typedef __attribute__((ext_vector_type(4))) int v4i;

#if defined(__has_builtin)
#if __has_builtin(__builtin_amdgcn_tensor_load_to_lds) && __has_builtin(__builtin_amdgcn_s_wait_tensorcnt)
#define HAVE_TDM 1
#endif
#endif

#ifdef HAVE_TDM
// 2D tile load Global->LDS via Tensor Data Mover.
// dims/tiles/strides in elements (data_size = 4 bytes). Optional LDS padding:
// pad_int_code: ISA code (6 => every 128 dwords), pad_amt_code: amount-1 dwords.
__device__ __forceinline__ void tdm_load_2d(unsigned lds_addr, const void* gptr,
                                            unsigned dim0, unsigned dim1,
                                            unsigned tile0, unsigned tile1,
                                            unsigned long long stride0,
                                            unsigned pad_int_code, unsigned pad_amt_code,
                                            int pad_enable) {
    unsigned long long ga = (unsigned long long)(uintptr_t)gptr;
    v4u g0;
    g0[0] = 1u;                                                // count=1, user descriptor
    g0[1] = lds_addr;                                          // LDS byte address
    g0[2] = (unsigned)(ga & 0xFFFFFFFFu);                      // global_addr[31:0]
    g0[3] = (unsigned)((ga >> 32) & 0x01FFFFFFu) | (2u << 30); // global_addr[56:32] | type=2
    v8i g1;
    unsigned d0 = (2u << 16);                                  // data_size=2 -> 4 bytes
    if (pad_enable) d0 |= (1u << 20) | (pad_int_code << 22) | (pad_amt_code << 25);
    g1[0] = (int)d0;
    g1[1] = (int)((dim0 & 0xFFFFu) << 16);                     // tensor_dim0[15:0]
    g1[2] = (int)((dim0 >> 16) | ((dim1 & 0xFFFFu) << 16));    // dim0[31:16] | dim1[15:0]
    g1[3] = (int)((dim1 >> 16) | (tile0 << 16));               // dim1[31:16] | tile_dim0
    g1[4] = (int)(tile1 & 0xFFFFu);                            // tile_dim1 | tile_dim2=0
    g1[5] = (int)(stride0 & 0xFFFFFFFFu);                      // tensor_dim0_stride[31:0]
    g1[6] = (int)((stride0 >> 32) & 0xFFFFu);                  // stride0[47:32] | stride1=0
    g1[7] = 0;
    v4i z = {0, 0, 0, 0};
#if __clang_major__ >= 23
    v8i z8 = {0, 0, 0, 0, 0, 0, 0, 0};
    __builtin_amdgcn_tensor_load_to_lds(g0, g1, z, z, z8, 0);
#else
    __builtin_amdgcn_tensor_load_to_lds(g0, g1, z, z, 0);
#endif
}
#endif

// ---------------- utility ----------------
__global__ void zero_kernel(float* __restrict__ p, size_t n) {
    size_t i = (size_t)blockIdx.x * blockDim.x + threadIdx.x;
    size_t stride = (size_t)gridDim.x * blockDim.x;
    for (; i < n; i += stride) p[i] = 0.0f;
}

// ---------------- substructure mean accumulation ----------------
__global__ void subst_accum_kernel(const float* __restrict__ x, const int* __restrict__ sid,
                                   float* __restrict__ sums, float* __restrict__ cnt) {
    __shared__ float s_sum[SS * DD];
    __shared__ float s_cnt[SS];
    for (int i = threadIdx.x; i < SS * DD; i += blockDim.x) s_sum[i] = 0.0f;
    for (int i = threadIdx.x; i < SS; i += blockDim.x) s_cnt[i] = 0.0f;
    __syncthreads();

    int wave = threadIdx.x >> 5;
    int lane = threadIdx.x & 31;
    int wpb  = blockDim.x >> 5;
    for (int n = blockIdx.x * wpb + wave; n < NN; n += gridDim.x * wpb) {
        int s = sid[n];
        const float4* xr = (const float4*)(x + (size_t)n * DD);
        float4 v = xr[lane];
        atomicAdd(&s_sum[s * DD + lane * 4 + 0], v.x);
        atomicAdd(&s_sum[s * DD + lane * 4 + 1], v.y);
        atomicAdd(&s_sum[s * DD + lane * 4 + 2], v.z);
        atomicAdd(&s_sum[s * DD + lane * 4 + 3], v.w);
        if (lane == 0) atomicAdd(&s_cnt[s], 1.0f);
    }
    __syncthreads();
    for (int i = threadIdx.x; i < SS * DD; i += blockDim.x) atomicAdd(&sums[i], s_sum[i]);
    for (int i = threadIdx.x; i < SS; i += blockDim.x) atomicAdd(&cnt[i], s_cnt[i]);
}

// ---------------- attention scores + softmax (1 block, 64 threads) ----------------
__global__ void attn_kernel(const float* __restrict__ sums, const float* __restrict__ cnt,
                            const float* __restrict__ w1, const float* __restrict__ b1,
                            const float* __restrict__ w2, float* __restrict__ sattn) {
    __shared__ float sh[SS];
    __shared__ float mx, denom;
    int s = threadIdx.x;
    float inv = 1.0f / fmaxf(cnt[s], 1.0f);
    float score = 0.0f;
    for (int j = 0; j < 64; ++j) {
        float acc = b1[j];
        for (int d = 0; d < DD; ++d) acc += sums[s * DD + d] * inv * w1[d * 64 + j];
        score += tanhf(acc) * w2[j];
    }
    sh[s] = score;
    __syncthreads();
    if (s == 0) {
        float m = sh[0];
        for (int i = 1; i < SS; ++i) m = fmaxf(m, sh[i]);
        mx = m;
    }
    __syncthreads();
    float e = expf(score - mx);
    sh[s] = e;
    __syncthreads();
    if (s == 0) {
        float t = 0.0f;
        for (int i = 0; i < SS; ++i) t += sh[i];
        denom = t;
    }
    __syncthreads();
    sattn[s] = e / denom;
}

// ---------------- degree / symmetric norm ----------------
__global__ void deg_kernel(const int* __restrict__ dst, float* __restrict__ deg) {
    int i = blockIdx.x * blockDim.x + threadIdx.x;
    if (i < EE) atomicAdd(&deg[dst[i]], 1.0f);
}
__global__ void dinv_kernel(float* __restrict__ deg) {
    int i = blockIdx.x * blockDim.x + threadIdx.x;
    if (i < NN) deg[i] = rsqrtf(deg[i] + 1.0f);
}

// ---------------- WMMA fp32 GEMM, TDM double-buffered pipeline ----------------
// Persistent blocks. W (128x128) TDM-loaded once; A tiles (16x128, padded rows)
// ping-pong between two LDS buffers: wave 0 issues the NEXT tile's TDM load,
// then waits tensorcnt<=1 so the prefetch stays in flight during compute
// (tensor ops complete in-order per wave, so <=1 outstanding implies the
// current tile -- and on iter 0 the W load -- are done).
template <bool HasAttn>
__global__ void gemm_kernel(const float* __restrict__ A, const float* __restrict__ W,
                            float* __restrict__ xw,
                            const float* __restrict__ wlast,
                            const float* __restrict__ sattn,
                            const int* __restrict__ sid,
                            int ntiles) {
    extern __shared__ float smem[];
    float* sW  = smem;                                 // 128*128
    float* sA0 = smem + HH * DD;                       // 16 * A_STRIDE
    float* sA1 = sA0 + 16 * A_STRIDE;                  // 16 * A_STRIDE

    int wave = threadIdx.x >> 5;
    int lane = threadIdx.x & 31;
    int half = lane >> 4;
    int l    = lane & 15;
    int colbase = wave * 16;

#ifdef HAVE_TDM
    if (threadIdx.x < 32) {
        tdm_load_2d((unsigned)(uintptr_t)sW, W, HH, HH, HH, HH, HH, 0, 0, 0);
        int t0 = blockIdx.x;
        if (t0 < ntiles)
            tdm_load_2d((unsigned)(uintptr_t)sA0, A + (size_t)t0 * 16 * DD,
                        DD, 16, DD, 16, DD, 6, 3, 1);
    }
#else
    for (int i = threadIdx.x; i < HH * DD; i += blockDim.x) sW[i] = W[i];
#endif

    int cur = 0;
    for (int tile = blockIdx.x; tile < ntiles; tile += gridDim.x) {
        float* sA = cur ? sA1 : sA0;
#ifdef HAVE_TDM
        if (threadIdx.x < 32) {
            int nxt = tile + gridDim.x;
            if (nxt < ntiles) {
                float* sAn = cur ? sA0 : sA1;          // buffer just vacated (post-barrier)
                tdm_load_2d((unsigned)(uintptr_t)sAn, A + (size_t)nxt * 16 * DD,
                            DD, 16, DD, 16, DD, 6, 3, 1);
                __builtin_amdgcn_s_wait_tensorcnt(1);  // current tile (and W) complete
            } else {
                __builtin_amdgcn_s_wait_tensorcnt(0);
            }
        }
#else
        __syncthreads();
        for (int i = threadIdx.x; i < 16 * DD; i += blockDim.x)
            sA[(i >> 7) * A_STRIDE + (i & 127)] = A[(size_t)tile * 16 * DD + i];
#endif
        __syncthreads();                               // staged data visible to all waves

        v8f c = {0.f, 0.f, 0.f, 0.f, 0.f, 0.f, 0.f, 0.f};
        const float* arow = sA + l * A_STRIDE;
#pragma unroll 8
        for (int k = 0; k < DD; k += 4) {
            int ks = k + 2 * half;                     // ISA 16x4 fp32 A/B lane layout
            v2f a, b;
            a.x = arow[ks];
            a.y = arow[ks + 1];
            b.x = sW[ks * HH + colbase + l];
            b.y = sW[(ks + 1) * HH + colbase + l];
            c = __builtin_amdgcn_wmma_f32_16x16x4_f32(
                    false, a, false, b, (short)0, c, false, false);
        }

        int rowbase = tile * 16;
        int n = colbase + l;
#pragma unroll
        for (int i = 0; i < 8; ++i) {
            int m = rowbase + i + 8 * half;            // C/D: VGPR i -> M=i / M=i+8
            float val = c[i];
            if (HasAttn) val += sattn[sid[m]] * wlast[n];
            xw[(size_t)m * HH + n] = val;
        }
        __syncthreads();                               // all reads done before buffer reuse
        cur ^= 1;
    }
}

// ---------------- edge scatter: agg[dst] += dinv[src]*dinv[dst] * xw[src] ----------------
__global__ void edge_kernel(const int* __restrict__ src, const int* __restrict__ dst,
                            const float* __restrict__ dinv, const float* __restrict__ xw,
                            float* __restrict__ agg) {
    int wave = threadIdx.x >> 5;
    int lane = threadIdx.x & 31;
    int e = blockIdx.x * (blockDim.x >> 5) + wave;
    if (e >= EE) return;
    int s = src[e], d = dst[e];
    float coef = dinv[s] * dinv[d];
    const float4* xs = (const float4*)(xw + (size_t)s * HH);
    float4 v = xs[lane];
    float* ag = agg + (size_t)d * HH + lane * 4;
    atomicAdd(ag + 0, coef * v.x);
    atomicAdd(ag + 1, coef * v.y);
    atomicAdd(ag + 2, coef * v.z);
    atomicAdd(ag + 3, coef * v.w);
}

// ---------------- self-loop + bias + BN(eval) + ELU/ReLU ----------------
__global__ void post_kernel(const float* __restrict__ agg, const float* __restrict__ xw,
                            const float* __restrict__ dinv, const float* __restrict__ bias,
                            const float* __restrict__ gma, const float* __restrict__ bta,
                            const float* __restrict__ rm, const float* __restrict__ rv,
                            int do_elu, float* __restrict__ hout) {
    int n = blockIdx.x;
    int j = threadIdx.x;
    size_t idx = (size_t)n * HH + j;
    float di = dinv[n];
    float v = agg[idx] + xw[idx] * di * di + bias[j];
    v = (v - rm[j]) * rsqrtf(rv[j] + EPSV) * gma[j] + bta[j];
    v = do_elu ? (v > 0.0f ? v : expm1f(v)) : fmaxf(v, 0.0f);
    hout[idx] = v;
}

// ---------------- global mean pool ----------------
__global__ void pool_kernel(const float* __restrict__ h, const int* __restrict__ batch,
                            float* __restrict__ pooled, float* __restrict__ gcnt) {
    int n = blockIdx.x;
    int j = threadIdx.x;
    int b = batch[n];
    atomicAdd(&pooled[(size_t)b * HH + j], h[(size_t)n * HH + j]);
    if (j == 0) atomicAdd(&gcnt[b], 1.0f);
}

// ---------------- prediction head (per graph) ----------------
__global__ void head_kernel(const float* __restrict__ pooled, const float* __restrict__ gcnt,
                            const float* __restrict__ w1, const float* __restrict__ b1,
                            const float* __restrict__ w2, const float* __restrict__ b2,
                            float* __restrict__ out) {
    __shared__ float sh[64];
    int g = blockIdx.x;
    int j = threadIdx.x;
    float inv = 1.0f / fmaxf(gcnt[g], 1.0f);
    float acc = b1[j];
    for (int d = 0; d < HH; ++d) acc += pooled[(size_t)g * HH + d] * inv * w1[d * 64 + j];
    float z = acc > 0.0f ? acc : expm1f(acc);
    sh[j] = z * w2[j];
    __syncthreads();
    for (int off = 32; off > 0; off >>= 1) {
        if (j < off) sh[j] += sh[j + off];
        __syncthreads();
    }
    if (j == 0) out[g] = sh[0] + b2[0];
}

extern "C" void kernel_launch(void* const* d_in, const int* in_sizes, int n_in,
                              void* d_out, int out_size, void* d_ws, size_t ws_size,
                              hipStream_t stream) {
    const float* x        = (const float*)d_in[0];
    const int*   edge     = (const int*)d_in[1];
    const int*   sid      = (const int*)d_in[2];
    const int*   batch    = (const int*)d_in[3];
    const float* attn_w1  = (const float*)d_in[4];
    const float* attn_b1  = (const float*)d_in[5];
    const float* attn_w2  = (const float*)d_in[6];
    const float* conv_w[4] = {(const float*)d_in[7],  (const float*)d_in[9],
                              (const float*)d_in[11], (const float*)d_in[13]};
    const float* conv_b[4] = {(const float*)d_in[8],  (const float*)d_in[10],
                              (const float*)d_in[12], (const float*)d_in[14]};
    const float* bn_gamma = (const float*)d_in[15];
    const float* bn_beta  = (const float*)d_in[16];
    const float* bn_rmean = (const float*)d_in[17];
    const float* bn_rvar  = (const float*)d_in[18];
    const float* head_w1  = (const float*)d_in[19];
    const float* head_b1  = (const float*)d_in[20];
    const float* head_w2  = (const float*)d_in[21];
    const float* head_b2  = (const float*)d_in[22];
    float* out = (float*)d_out;

    float* ws = (float*)d_ws;
    size_t o = 0;
    float* hcur   = ws + o; o += (size_t)NN * HH;
    float* xw     = ws + o; o += (size_t)NN * HH;
    float* agg    = ws + o; o += (size_t)NN * HH;
    float* dinv   = ws + o; o += NN;
    float* sums   = ws + o; o += SS * DD;
    float* cnt    = ws + o; o += SS;
    float* sattn  = ws + o; o += SS;
    float* pooled = ws + o; o += (size_t)GG * HH;
    float* gcnt   = ws + o; o += GG;

    const int* srcv = edge;
    const int* dstv = edge + EE;

    // zero accumulator regions
    zero_kernel<<<256, 256, 0, stream>>>(dinv, (size_t)NN);
    zero_kernel<<<64, 256, 0, stream>>>(sums, (size_t)(SS * DD + SS + SS));
    zero_kernel<<<256, 256, 0, stream>>>(pooled, (size_t)GG * HH + GG);

    // substructure attention
    subst_accum_kernel<<<512, 256, 0, stream>>>(x, sid, sums, cnt);
    attn_kernel<<<1, 64, 0, stream>>>(sums, cnt, attn_w1, attn_b1, attn_w2, sattn);

    // degree -> d^-1/2 (self-loops included)
    deg_kernel<<<(EE + 255) / 256, 256, 0, stream>>>(dstv, dinv);
    dinv_kernel<<<(NN + 255) / 256, 256, 0, stream>>>(dinv);

    const int ntiles = NN / 16;                       // 6250 (exact)
    const size_t smem = (size_t)(HH * DD + 2 * 16 * A_STRIDE) * sizeof(float);

    for (int layer = 0; layer < 4; ++layer) {
        const float* Ain = (layer == 0) ? x : hcur;
        if (layer == 0) {
            gemm_kernel<true><<<1024, 256, smem, stream>>>(
                Ain, conv_w[0], xw, conv_w[0] + 128 * HH, sattn, sid, ntiles);
        } else {
            gemm_kernel<false><<<1024, 256, smem, stream>>>(
                Ain, conv_w[layer], xw, nullptr, nullptr, nullptr, ntiles);
        }
        zero_kernel<<<2048, 256, 0, stream>>>(agg, (size_t)NN * HH);
        edge_kernel<<<(EE + 7) / 8, 256, 0, stream>>>(srcv, dstv, dinv, xw, agg);
        post_kernel<<<NN, HH, 0, stream>>>(agg, xw, dinv, conv_b[layer],
                                           bn_gamma + layer * HH, bn_beta + layer * HH,
                                           bn_rmean + layer * HH, bn_rvar + layer * HH,
                                           (layer < 3) ? 1 : 0, hcur);
    }

    pool_kernel<<<NN, HH, 0, stream>>>(hcur, batch, pooled, gcnt);
    head_kernel<<<GG, 64, 0, stream>>>(pooled, gcnt, head_w1, head_b1, head_w2, head_b2, out);
}